// StructNConv2D_s_with_d_69114613727826
// MI455X (gfx1250) — compile-verified
//
#include <hip/hip_runtime.h>

typedef __attribute__((ext_vector_type(2))) float v2f;
typedef __attribute__((ext_vector_type(8))) float v8f;

#define EPSF   1e-20f
#define NEGF  (-1e30f)

constexpr int Bn  = 2;
constexpr int Cn  = 16;
constexpr int Hn  = 352;
constexpr int Wn  = 1216;
constexpr int Hon = 176;                      // (352 + 2 - 3)/2 + 1
constexpr int Won = 608;                      // (1216 + 2 - 3)/2 + 1
constexpr int OUT_PLANE = Bn * Cn * Hon * Won; // elems per output tensor

// ---------------------------------------------------------------------------
// Pre-pass: scalar sums of spatial_weight (144 elems) and channel_weight (256)
// ---------------------------------------------------------------------------
__global__ __launch_bounds__(256) void weight_sums_kernel(
    const float* __restrict__ cwg, const float* __restrict__ spwg,
    float* __restrict__ sums) {
  __shared__ float rSp[256];
  __shared__ float rCw[256];
  const int t = threadIdx.x;
  rSp[t] = (t < Cn * 9) ? spwg[t] : 0.0f;
  rCw[t] = cwg[t];
  __syncthreads();
  for (int off = 128; off > 0; off >>= 1) {
    if (t < off) { rSp[t] += rSp[t + off]; rCw[t] += rCw[t + off]; }
    __syncthreads();
  }
  if (t == 0) { sums[0] = rSp[0]; sums[1] = rCw[0]; }
}

// ---------------------------------------------------------------------------
// Fused kernel: patch stage (argmax gathers + depthwise conv) -> LDS ->
// 16x16 channel matmul on V_WMMA_F32_16X16X4_F32 -> final fusion -> stores.
// Block = 256 threads = one (16 channel x 16 pixel) tile; wave 0 runs WMMA.
// ---------------------------------------------------------------------------
__global__ __launch_bounds__(256) void fused_kernel(
    const float* __restrict__ dg,  const float* __restrict__ cdg,
    const float* __restrict__ sg,  const float* __restrict__ csg,
    const float* __restrict__ wsdg, const float* __restrict__ wprg,
    const float* __restrict__ cwg,  const float* __restrict__ spwg,
    const float* __restrict__ sums, float* __restrict__ outg) {
  __shared__ float lds_cw[16][16];   // channel_weight[o][i]
  __shared__ float lds_x1[16][16];   // cs1*s1   [channel][pixel]
  __shared__ float lds_x2[16][16];   // cs1      [channel][pixel]
  __shared__ float lds_sfd[16][16];  // s_from_d [channel][pixel]
  __shared__ float lds_csfd[16][16]; // cs_from_d[channel][pixel]
  __shared__ float lds_spw[16 * 9];
  __shared__ float lds_wprop[16];
  __shared__ float lds_wsd[16];

  const int tid = threadIdx.x;
  const int p   = tid & 15;   // pixel within tile (wo offset)
  const int c   = tid >> 4;   // channel
  const int ho  = blockIdx.y;
  const int b   = blockIdx.z;
  const int woB = blockIdx.x * 16;
  const int wo  = woB + p;

  // ---- stage small weights into LDS ----
  lds_cw[c][p] = cwg[tid];                    // flat o*16+i == tid
  if (tid < Cn * 9) lds_spw[tid] = spwg[tid];
  if (tid < 16) { lds_wprop[tid] = wprg[tid]; lds_wsd[tid] = wsdg[tid]; }
  const float sumSp = sums[0];
  const float sumCw = sums[1];
  __syncthreads();

  // ---- patch stage: this thread handles (channel c, pixel p) ----
  const int plane = Hn * Wn;
  const int base  = (b * Cn + c) * plane;
  const float* dP  = dg  + base;
  const float* cdP = cdg + base;
  const float* sP  = sg  + base;
  const float* csP = csg + base;

  float bestMax = NEGF, bestMin = NEGF;
  float dMax = 0.f, cdMax = 0.f, dMin = 0.f, cdMin = 0.f;
  float nom = 0.f, den = 0.f;
  const int hi0 = ho * 2 - 1;
  const int wi0 = wo * 2 - 1;
#pragma unroll
  for (int ki = 0; ki < 3; ++ki) {
    const int  hi  = hi0 + ki;
    const bool hOK = (hi >= 0) && (hi < Hn);
#pragma unroll
    for (int kj = 0; kj < 3; ++kj) {
      const int  wi = wi0 + kj;
      const bool ok = hOK && (wi >= 0) && (wi < Wn);
      float dv = 0.f, cdv = 0.f, sv = 0.f, csv = 0.f;
      if (ok) {
        const int off = hi * Wn + wi;
        dv  = dP[off];  cdv = cdP[off];
        sv  = sP[off];  csv = csP[off];
      }
      // argmax criteria (first-max wins, matching jnp.argmax tie-break; pads = NEG)
      const float critMax = ok ? dv * cdv : NEGF;
      const float critMin = ok ? cdv / (dv + EPSF) : NEGF;
      if (critMax > bestMax) { bestMax = critMax; dMax = dv; cdMax = cdv; }
      if (critMin > bestMin) { bestMin = critMin; dMin = dv; cdMin = cdv; }
      // depthwise conv (zero padding: csv/sv already 0 when !ok)
      const float w = lds_spw[c * 9 + ki * 3 + kj];
      nom += w * (csv * sv);
      den += w * csv;
    }
  }

  const float mdm  = dMin / (dMax + EPSF);
  const float wsd  = lds_wsd[c];
  const float sfd  = (1.0f - wsd) * mdm + wsd * (mdm * mdm);
  const float csfd = cdMax * cdMin;
  const float s1   = nom / (den + EPSF);
  const float cs1  = den / (sumSp + EPSF);

  lds_x1[c][p]   = cs1 * s1;
  lds_x2[c][p]   = cs1;
  lds_sfd[c][p]  = sfd;
  lds_csfd[c][p] = csfd;
  __syncthreads();

  // ---- wave 0: channel mixing via f32 WMMA, then final fusion + store ----
  if (tid < 32) {
    const int lane = tid;
    const int half = lane >> 4;  // ISA 16x16x4 f32 layout: lanes 16-31 carry K+2
    const int m    = lane & 15;  // A: M row index; B: N (pixel); D: N (pixel)

    v8f accN = {0.f, 0.f, 0.f, 0.f, 0.f, 0.f, 0.f, 0.f};
    v8f accD = {0.f, 0.f, 0.f, 0.f, 0.f, 0.f, 0.f, 0.f};
#pragma unroll
    for (int kc = 0; kc < 4; ++kc) {
      const int k0 = kc * 4 + half * 2;
      v2f a, bN, bD;
      a.x  = lds_cw[m][k0];      // A[M=m][K=k0]
      a.y  = lds_cw[m][k0 + 1];  // A[M=m][K=k0+1]
      bN.x = lds_x1[k0][m];      // B[K=k0][N=m]
      bN.y = lds_x1[k0 + 1][m];
      bD.x = lds_x2[k0][m];
      bD.y = lds_x2[k0 + 1][m];
      accN = __builtin_amdgcn_wmma_f32_16x16x4_f32(false, a, false, bN,
                                                   (short)0, accN, false, false);
      accD = __builtin_amdgcn_wmma_f32_16x16x4_f32(false, a, false, bD,
                                                   (short)0, accD, false, false);
    }
#pragma unroll
    for (int v = 0; v < 8; ++v) {
      const int   o    = v + half * 8;      // D layout: lanes 16-31 hold M=v+8
      const float nom2 = accN[v];
      const float den2 = accD[v];
      const float s2   = nom2 / (den2 + EPSF);
      const float cs2  = den2 / (sumCw + EPSF);
      const float wp   = lds_wprop[o];
      const float sv   = lds_sfd[o][m];
      const float cv   = lds_csfd[o][m];
      const float a1   = wp * cs2;
      const float s3   = (a1 * s2 + cv * sv) / (a1 + cv + EPSF);
      const float cs3  = (a1 + cv) / (wp + 1.0f);
      const int idx = ((b * Cn + o) * Hon + ho) * Won + woB + m;
      outg[idx]             = s3;
      outg[OUT_PLANE + idx] = cs3 * 0.25f;  // DEVALUE = 1/stride^2
    }
  }
}

// ---------------------------------------------------------------------------
extern "C" void kernel_launch(void* const* d_in, const int* in_sizes, int n_in,
                              void* d_out, int out_size, void* d_ws, size_t ws_size,
                              hipStream_t stream) {
  (void)in_sizes; (void)n_in; (void)out_size; (void)ws_size;
  const float* dg   = (const float*)d_in[0];
  const float* cdg  = (const float*)d_in[1];
  const float* sg   = (const float*)d_in[2];
  const float* csg  = (const float*)d_in[3];
  const float* wsdg = (const float*)d_in[4];
  const float* wprg = (const float*)d_in[5];
  const float* cwg  = (const float*)d_in[6];
  const float* spwg = (const float*)d_in[7];
  float* outg = (float*)d_out;
  float* sums = (float*)d_ws;

  weight_sums_kernel<<<1, 256, 0, stream>>>(cwg, spwg, sums);

  dim3 grid(Won / 16, Hon, Bn);   // (38, 176, 2)
  fused_kernel<<<grid, 256, 0, stream>>>(dg, cdg, sg, csg, wsdg, wprg,
                                         cwg, spwg, sums, outg);
}